// Attention_87892210745803
// MI455X (gfx1250) — compile-verified
//
#include <hip/hip_runtime.h>
#include <math.h>

// ---------------------------------------------------------------------------
// Fused QKV projection + causal multi-head attention for MI455X (gfx1250).
// fp32 end-to-end using V_WMMA_F32_16X16X4_F32 (exact fp32 matrix pipe).
// ---------------------------------------------------------------------------

typedef __attribute__((ext_vector_type(2))) float v2f;
typedef __attribute__((ext_vector_type(8))) float v8f;

#define EMB   1024
#define T3E   3072
#define TSEQ  2048
#define NBATCH 4
#define NH    16
#define HS    64

// ----------------------------------------------------------------------------
// Kernel 1: qkv[b*T+t, f] = sum_e x[b*T+t, e] * W[f, e] + bias[f]
// M = 8192, N = 3072, K = 1024.
//
// Each wave produces a 16x64 strip (4 accumulators) so the A fragment is
// loaded ONCE per K-step and reused by 4 WMMAs: 5 global_load_b64 per
// 4 v_wmma (1.25 vmem/wmma instead of 2.0). Waves in a block split M, so
// their 4 B streams hit the same L0/L2 lines (W is fully L2-resident).
//
// A 16x4 layout: lane l holds M = l%16; VGPR{0,1} = K {0,1} (lanes<16) or
// {2,3} (lanes>=16) -> contiguous float2 load from the x row.
// B 4x16 layout: lane l holds N = l%16; VGPR{0,1} = K rows {0,1}/{2,3} by
// half -> contiguous float2 load from the W row (W is [N,K] row-major, so
// B[r][n] = W[n][k+r]).
// C/D: M = vgpr + 8*half, N = lane%16.
// ----------------------------------------------------------------------------
__global__ __launch_bounds__(128) void qkv_gemm_kernel(
    const float* __restrict__ x, const float* __restrict__ W,
    const float* __restrict__ bias, float* __restrict__ qkv) {
  const int lane = threadIdx.x & 31;
  const int wave = threadIdx.x >> 5;
  const int half = lane >> 4;
  const int l16  = lane & 15;

  const int row0 = blockIdx.y * 64 + wave * 16;   // M tile (per wave)
  const int col0 = blockIdx.x * 64;               // N strip (4 x 16)

  const float* arow = x + (size_t)(row0 + l16) * EMB + 2 * half;
  const float* brow0 = W + (size_t)(col0 + l16) * EMB + 2 * half;
  const float* brow1 = brow0 + (size_t)16 * EMB;
  const float* brow2 = brow0 + (size_t)32 * EMB;
  const float* brow3 = brow0 + (size_t)48 * EMB;

  v8f acc[4];
#pragma unroll
  for (int n = 0; n < 4; ++n) {
    const float bv = bias[col0 + 16 * n + l16];  // per-N, same for all M
#pragma unroll
    for (int i = 0; i < 8; ++i) acc[n][i] = bv;
  }

#pragma unroll 4
  for (int k = 0; k < EMB; k += 4) {
    v2f a  = *(const v2f*)(arow + k);
    v2f b0 = *(const v2f*)(brow0 + k);
    v2f b1 = *(const v2f*)(brow1 + k);
    v2f b2 = *(const v2f*)(brow2 + k);
    v2f b3 = *(const v2f*)(brow3 + k);
    acc[0] = __builtin_amdgcn_wmma_f32_16x16x4_f32(false, a, false, b0,
                                                   (short)0, acc[0], false, false);
    acc[1] = __builtin_amdgcn_wmma_f32_16x16x4_f32(false, a, false, b1,
                                                   (short)0, acc[1], false, false);
    acc[2] = __builtin_amdgcn_wmma_f32_16x16x4_f32(false, a, false, b2,
                                                   (short)0, acc[2], false, false);
    acc[3] = __builtin_amdgcn_wmma_f32_16x16x4_f32(false, a, false, b3,
                                                   (short)0, acc[3], false, false);
  }

#pragma unroll
  for (int n = 0; n < 4; ++n) {
    float* out = qkv + (size_t)row0 * T3E + col0 + 16 * n + l16;
#pragma unroll
    for (int i = 0; i < 8; ++i)
      out[(size_t)(i + 8 * half) * T3E] = acc[n][i];
  }
}

// ----------------------------------------------------------------------------
// Kernel 2: flash-style causal attention, fp32 WMMA.
// qkv is [B*T, 3E]; for (b,h): q(t,d)=qkv[(b*T+t)*3E + h*64 + d],
// k at +E, v at +2E.  Output [B,T,E] with head h in cols h*64..h*64+63.
//
// One wave owns a 16-row query tile. Per 16-key tile:
//   S (16x16)  = Q K^T : 16 wmma (K-dim 64 in steps of 4), Q register-resident
//   online softmax over rows (row = (vgpr, half); reduce over 16 lanes/half)
//   P  C-layout -> A-layout via per-wave LDS tile (stride 17, s_wait_dscnt)
//   O (16x64) += P V    : 4 N-tiles x 4 K-steps = 16 wmma
// ----------------------------------------------------------------------------
__global__ __launch_bounds__(128) void attn_kernel(
    const float* __restrict__ qkv, float* __restrict__ out) {
  __shared__ float pls[4][16 * 17];   // per-wave P tile, padded stride 17

  const int lane = threadIdx.x & 31;
  const int wave = threadIdx.x >> 5;
  const int half = lane >> 4;
  const int l16  = lane & 15;

  const int bh = blockIdx.y;          // 0..63
  const int b  = bh >> 4;
  const int h  = bh & 15;
  const int qt = blockIdx.x * 4 + wave;  // query tile 0..127
  const int qrow = qt * 16;

  const float* qbase = qkv + (size_t)b * TSEQ * T3E + (size_t)h * HS;
  const float* kbase = qbase + EMB;
  const float* vbase = qbase + 2 * EMB;

  // Preload this wave's Q tile in A-layout registers: per k-step ks,
  // lane holds Q[qrow + l16][4*ks + 2*half .. +1].
  v2f qreg[16];
  {
    const float* qp = qbase + (size_t)(qrow + l16) * T3E + 2 * half;
#pragma unroll
    for (int ks = 0; ks < 16; ++ks) qreg[ks] = *(const v2f*)(qp + ks * 4);
  }

  v8f o[4];
#pragma unroll
  for (int nb = 0; nb < 4; ++nb)
#pragma unroll
    for (int i = 0; i < 8; ++i) o[nb][i] = 0.0f;

  float mrow[8], lrow[8];
#pragma unroll
  for (int i = 0; i < 8; ++i) { mrow[i] = -__builtin_inff(); lrow[i] = 0.0f; }

  float* plds = pls[wave];
  const float scale = 0.125f;   // 1/sqrt(64)

  for (int jt = 0; jt <= qt; ++jt) {
    const int krow = jt * 16;
    const float* kp = kbase + (size_t)(krow + l16) * T3E + 2 * half;

    // Prefetch next key tile row into the cache hierarchy.
    if (jt < qt)
      __builtin_prefetch((const void*)(kp + 16 * T3E), 0, 1);

    // S = Q * K^T  (B[r][n] = K[krow+n][d0+r] -> float2 from K row)
    v8f s;
#pragma unroll
    for (int i = 0; i < 8; ++i) s[i] = 0.0f;
#pragma unroll
    for (int ks = 0; ks < 16; ++ks) {
      v2f bk = *(const v2f*)(kp + ks * 4);
      s = __builtin_amdgcn_wmma_f32_16x16x4_f32(
          false, qreg[ks], false, bk, (short)0, s, false, false);
    }

    // Scale + causal mask (only the diagonal tile needs masking).
    float p[8];
#pragma unroll
    for (int i = 0; i < 8; ++i) {
      float v = s[i] * scale;
      if (jt == qt && (krow + l16) > (qrow + i + 8 * half))
        v = -__builtin_inff();
      p[i] = v;
    }

    // Online softmax. Row M=(i + 8*half) lives in 16 lanes of this half;
    // xor-shuffles with masks 1..8 stay inside the half.
#pragma unroll
    for (int i = 0; i < 8; ++i) {
      float r = p[i];
      r = fmaxf(r, __shfl_xor(r, 1, 32));
      r = fmaxf(r, __shfl_xor(r, 2, 32));
      r = fmaxf(r, __shfl_xor(r, 4, 32));
      r = fmaxf(r, __shfl_xor(r, 8, 32));
      const float mnew  = fmaxf(mrow[i], r);       // always finite (N=0 unmasked)
      const float alpha = __expf(mrow[i] - mnew);  // exp(-inf)=0 on first tile
      const float e     = __expf(p[i] - mnew);
      float rs = e;
      rs += __shfl_xor(rs, 1, 32);
      rs += __shfl_xor(rs, 2, 32);
      rs += __shfl_xor(rs, 4, 32);
      rs += __shfl_xor(rs, 8, 32);
      lrow[i] = lrow[i] * alpha + rs;
      mrow[i] = mnew;
      p[i] = e;
#pragma unroll
      for (int nb = 0; nb < 4; ++nb) o[nb][i] *= alpha;
    }

    // P: C-layout -> A-layout via LDS (row-major 16x16 tile, stride 17).
#pragma unroll
    for (int i = 0; i < 8; ++i)
      plds[(i + 8 * half) * 17 + l16] = p[i];
    asm volatile("s_wait_dscnt 0" ::: "memory");  // intra-wave LDS RAW

    v2f pa[4];
#pragma unroll
    for (int k2 = 0; k2 < 4; ++k2)
      pa[k2] = *(const v2f*)(plds + l16 * 17 + k2 * 4 + 2 * half);

    // O += P * V.  B[r][n] = V[krow + 4*k2 + r][nb*16 + n]:
    // b.x = row (4*k2 + 2*half), b.y = row (4*k2 + 2*half + 1).
#pragma unroll
    for (int nb = 0; nb < 4; ++nb) {
#pragma unroll
      for (int k2 = 0; k2 < 4; ++k2) {
        const float* vp =
            vbase + (size_t)(krow + k2 * 4 + 2 * half) * T3E + nb * 16 + l16;
        v2f bv2;
        bv2.x = vp[0];
        bv2.y = vp[T3E];
        o[nb] = __builtin_amdgcn_wmma_f32_16x16x4_f32(
            false, pa[k2], false, bv2, (short)0, o[nb], false, false);
      }
    }
  }

  // Normalize and store: out[(b*T + qrow + M)*E + h*64 + nb*16 + N]
  float* op = out + (size_t)(b * TSEQ + qrow) * EMB + h * HS;
#pragma unroll
  for (int i = 0; i < 8; ++i) {
    const float inv = 1.0f / lrow[i];
#pragma unroll
    for (int nb = 0; nb < 4; ++nb)
      op[(size_t)(i + 8 * half) * EMB + nb * 16 + l16] = o[nb][i] * inv;
  }
}

// ----------------------------------------------------------------------------
extern "C" void kernel_launch(void* const* d_in, const int* in_sizes, int n_in,
                              void* d_out, int out_size, void* d_ws, size_t ws_size,
                              hipStream_t stream) {
  const float* x    = (const float*)d_in[0];  // [4, 2048, 1024]
  const float* Wqkv = (const float*)d_in[1];  // [3072, 1024]
  const float* bqkv = (const float*)d_in[2];  // [3072]
  float* out = (float*)d_out;                 // [4, 2048, 1024]
  float* qkv = (float*)d_ws;                  // [8192, 3072] fp32 = 96 MB

  // QKV GEMM: grid (3072/64, 8192/64) = (48, 128), 4 waves/block,
  // wave computes a 16x64 strip.
  dim3 g1(T3E / 64, (NBATCH * TSEQ) / 64);
  qkv_gemm_kernel<<<g1, 128, 0, stream>>>(x, Wqkv, bqkv, qkv);

  // Attention: grid (128 qtiles / 4 waves, B*H) = (32, 64), 4 waves/block.
  dim3 g2(TSEQ / 16 / 4, NBATCH * NH);
  attn_kernel<<<g2, 128, 0, stream>>>(qkv, out);
}